// Memory_6554120093948
// MI455X (gfx1250) — compile-verified
//
#include <hip/hip_runtime.h>

#define C_    1000
#define D_    256
#define B_    16384
#define NS_   200000
#define CPAD  1008
#define NTILE 63
#define MOM   0.1f
#define EPS_  1e-12f

typedef __attribute__((ext_vector_type(2))) float v2f;
typedef __attribute__((ext_vector_type(8))) float v8f;

// ---------------- row normalize (one block of 256 threads per row) ----------
__global__ __launch_bounds__(256) void normalize_rows_kernel(
    const float* __restrict__ src, float* __restrict__ dst, int nrows_valid)
{
    __shared__ float red[256];
    const int r = blockIdx.x;
    const int t = threadIdx.x;
    if (r >= nrows_valid) {            // padded rows: write zeros
        dst[(size_t)r * D_ + t] = 0.0f;
        return;
    }
    float v = src[(size_t)r * D_ + t];
    red[t] = v * v;
    __syncthreads();
    for (int s = 128; s > 0; s >>= 1) {
        if (t < s) red[t] += red[t + s];
        __syncthreads();
    }
    float nrm = sqrtf(red[0]);
    float scale = 1.0f / fmaxf(nrm, EPS_);
    dst[(size_t)r * D_ + t] = v * scale;
}

// ---------------- dist_at_label[i] = 0.5*(1 - fn[i].cn1[label[i]]) ----------
__global__ __launch_bounds__(256) void dal_kernel(
    const float* __restrict__ fn, const float* __restrict__ cn1,
    const int* __restrict__ label, float* __restrict__ dal)
{
    __shared__ float red[256];
    const int i = blockIdx.x;
    const int t = threadIdx.x;
    const int lbl = label[i];
    red[t] = fn[(size_t)i * D_ + t] * cn1[(size_t)lbl * D_ + t];
    __syncthreads();
    for (int s = 128; s > 0; s >>= 1) {
        if (t < s) red[t] += red[t + s];
        __syncthreads();
    }
    if (t == 0) dal[i] = 0.5f * (1.0f - red[0]);
}

// ---- per-class deterministic segment sum + class2center (block per class) --
__global__ __launch_bounds__(256) void segsum_kernel(
    const float* __restrict__ feat, const float* __restrict__ cmem,
    const float* __restrict__ c2c_in, const int* __restrict__ label,
    const float* __restrict__ dal,
    float* __restrict__ out_center, float* __restrict__ out_c2c)
{
    const int c = blockIdx.x;
    const int d = threadIdx.x;
    float acc = 0.0f, cnt = 0.0f, dsum = 0.0f;
    for (int i = 0; i < B_; ++i) {
        if (label[i] == c) {                    // uniform -> scalar branch
            acc += feat[(size_t)i * D_ + d];
            if (d == 0) { cnt += 1.0f; dsum += dal[i]; }
        }
    }
    out_center[(size_t)c * D_ + d] = cmem[(size_t)c * D_ + d] + acc;
    if (d == 0) {
        out_c2c[2 * c + 0] = c2c_in[2 * c + 0] + cnt;
        out_c2c[2 * c + 1] = c2c_in[2 * c + 1] + dsum;
    }
}

// ---------------- plain grid-stride copy ------------------------------------
__global__ __launch_bounds__(256) void copy_kernel(
    const float* __restrict__ src, float* __restrict__ dst, size_t n)
{
    size_t j = (size_t)blockIdx.x * blockDim.x + threadIdx.x;
    size_t stride = (size_t)gridDim.x * blockDim.x;
    for (; j < n; j += stride) dst[j] = src[j];
}

// ---------------- momentum blend of selected rows ---------------------------
__global__ __launch_bounds__(256) void blend_kernel(
    const float* __restrict__ smem, const float* __restrict__ feat,
    const int* __restrict__ index_source, float* __restrict__ out_src)
{
    size_t j = (size_t)blockIdx.x * blockDim.x + threadIdx.x;
    if (j >= (size_t)B_ * D_) return;
    int i = (int)(j / D_);
    int d = (int)(j % D_);
    int r = index_source[i];
    out_src[(size_t)r * D_ + d] =
        (1.0f - MOM) * smem[(size_t)r * D_ + d] + MOM * feat[j];
}

// ------ WMMA f32 16x16x4 GEMM (fn @ cn2^T) fused with per-row top-2 --------
__global__ __launch_bounds__(256) void wmma_topk_kernel(
    const float* __restrict__ fn, const float* __restrict__ cn2,
    float* __restrict__ out_inter)
{
    __shared__ float tile[8][16][17];          // per-wave 16x16 distance tile
    const int lane = threadIdx.x & 31;
    const int wave = threadIdx.x >> 5;
    const int half = lane >> 4;                // K sub-block select (A & B)
    const int mr   = lane & 15;                // M index for A, N index for B
    const int rowBase = (blockIdx.x * 8 + wave) * 16;

    const float* arow = fn + (size_t)(rowBase + mr) * D_ + 2 * half;

    float td0 = 3.0e38f, td1 = 3.0e38f;
    int   ti0 = 0,       ti1 = 0;

    for (int t = 0; t < NTILE; ++t) {
        const int colBase = t * 16;
        const float* brow = cn2 + (size_t)(colBase + mr) * D_ + 2 * half;

        v8f acc = {};
        #pragma unroll 8
        for (int kk = 0; kk < D_; kk += 4) {
            v2f a = *(const v2f*)(arow + kk);   // A: 16x4 f32 fragment
            v2f b = *(const v2f*)(brow + kk);   // B: 4x16 f32 fragment
            acc = __builtin_amdgcn_wmma_f32_16x16x4_f32(
                false, a, false, b, (short)0, acc, false, false);
        }

        const bool colOK = (colBase + mr) < C_;
        #pragma unroll
        for (int v = 0; v < 8; ++v) {
            float dval = colOK ? 0.5f * (1.0f - acc[v]) : 3.0e38f;
            tile[wave][v + 8 * half][mr] = dval;   // D layout: (M=v+8*half, N=mr)
        }
        __syncthreads();
        if (lane < 16) {                       // lane == row within tile
            #pragma unroll
            for (int n = 0; n < 16; ++n) {
                float d = tile[wave][lane][n];
                int idx = colBase + n;
                if (d < td1) {                 // strict < keeps earliest index
                    if (d < td0) { td1 = td0; ti1 = ti0; td0 = d; ti0 = idx; }
                    else         { td1 = d;   ti1 = idx; }
                }
            }
        }
        __syncthreads();
    }

    if (lane < 16) {
        float inter = fabsf(td0 - td1);
        float* p = out_inter + ((size_t)ti0 * C_ + ti1) * 2;
        atomicAdd(p + 0, 1.0f);
        atomicAdd(p + 1, inter);
    }
}

extern "C" void kernel_launch(void* const* d_in, const int* in_sizes, int n_in,
                              void* d_out, int out_size, void* d_ws, size_t ws_size,
                              hipStream_t stream)
{
    (void)in_sizes; (void)n_in; (void)out_size; (void)ws_size;

    const float* feat  = (const float*)d_in[0];   // B x D
    const float* cmem  = (const float*)d_in[1];   // C x D
    const float* smem  = (const float*)d_in[2];   // NS x D
    const float* c2c   = (const float*)d_in[3];   // C x 2
    const float* inter = (const float*)d_in[4];   // C x C x 2
    const int*   label = (const int*)d_in[5];     // B
    const int*   index = (const int*)d_in[6];     // B

    float* out        = (float*)d_out;
    float* out_center = out;                                  // C*D
    float* out_src    = out_center + (size_t)C_ * D_;         // NS*D
    float* out_c2c    = out_src + (size_t)NS_ * D_;           // C*2
    float* out_inter  = out_c2c + (size_t)C_ * 2;             // C*C*2

    float* ws  = (float*)d_ws;
    float* fn  = ws;                                          // B*D
    float* cn1 = fn + (size_t)B_ * D_;                        // C*D
    float* cn2 = cn1 + (size_t)C_ * D_;                       // CPAD*D
    float* dal = cn2 + (size_t)CPAD * D_;                     // B

    // 1) normalize features and old centers
    normalize_rows_kernel<<<B_, 256, 0, stream>>>(feat, fn, B_);
    normalize_rows_kernel<<<C_, 256, 0, stream>>>(cmem, cn1, C_);

    // 2) dist at label
    dal_kernel<<<B_, 256, 0, stream>>>(fn, cn1, label, dal);

    // 3) center_new + class2center_new (deterministic per-class scan)
    segsum_kernel<<<C_, 256, 0, stream>>>(feat, cmem, c2c, label, dal,
                                          out_center, out_c2c);

    // 4) normalize updated centers, padded to 1008 rows (zeros past C)
    normalize_rows_kernel<<<CPAD, 256, 0, stream>>>(out_center, cn2, C_);

    // 5) source memory copy + momentum blend; interdist base copy
    {
        size_t n = (size_t)NS_ * D_;
        copy_kernel<<<(int)((n + 255) / 256), 256, 0, stream>>>(smem, out_src, n);
        blend_kernel<<<(B_ * D_) / 256, 256, 0, stream>>>(smem, feat, index, out_src);
        size_t ni = (size_t)C_ * C_ * 2;
        copy_kernel<<<(int)((ni + 255) / 256), 256, 0, stream>>>(inter, out_inter, ni);
    }

    // 6) fused WMMA distance GEMM + top-2 + interdist scatter
    wmma_topk_kernel<<<(B_ / 16) / 8, 256, 0, stream>>>(fn, cn2, out_inter);
}